// NMSLayer_54391465837297
// MI455X (gfx1250) — compile-verified
//
#include <hip/hip_runtime.h>
#include <stdint.h>

#define H1_BITS 14
#define H1_BINS (1u << H1_BITS)          // 16384 bins, 64KB LDS
#define H2_BITS 18
#define H2_BINS (1u << H2_BITS)          // 262144 bins, 1MB (lives in d_out scratch)

#define W 3072
#define TW 128
#define TH 32
#define HALO 3
#define LHW 4                             // left column halo (4 => 4-aligned b128 groups)
#define LW 136                            // tile cols: gcol = col0 - 4 + c, c in [0,136)
#define NG (LW / 4)                       // 34 b128 groups per row
#define LH (TH + 2*HALO)                  // 38 rows

__device__ __forceinline__ uint32_t f2key(float f) {
    uint32_t b = __float_as_uint(f);
    return (b & 0x80000000u) ? ~b : (b | 0x80000000u);  // monotonic total order
}

// ---------------- zero scratch ----------------
__global__ void zero_kernel(uint32_t* __restrict__ h, uint32_t* __restrict__ ws, int n) {
    int i = blockIdx.x * blockDim.x + threadIdx.x;
    if (i < n) h[i] = 0u;
    if (i < 8) ws[i] = 0u;
}

// ---------------- pass 1: histogram of top 14 key bits ----------------
__global__ __launch_bounds__(1024) void hist1_kernel(const float4* __restrict__ x, long nv,
                                                     uint32_t* __restrict__ g) {
    __shared__ uint32_t sh[H1_BINS];
    for (int i = threadIdx.x; i < (int)H1_BINS; i += blockDim.x) sh[i] = 0u;
    __syncthreads();
    long stride = (long)gridDim.x * blockDim.x;
    for (long i = (long)blockIdx.x * blockDim.x + threadIdx.x; i < nv; i += stride) {
        float4 v = x[i];
        atomicAdd(&sh[f2key(v.x) >> (32 - H1_BITS)], 1u);
        atomicAdd(&sh[f2key(v.y) >> (32 - H1_BITS)], 1u);
        atomicAdd(&sh[f2key(v.z) >> (32 - H1_BITS)], 1u);
        atomicAdd(&sh[f2key(v.w) >> (32 - H1_BITS)], 1u);
    }
    __syncthreads();
    for (int i = threadIdx.x; i < (int)H1_BINS; i += blockDim.x) {
        uint32_t c = sh[i];
        if (c) atomicAdd(&g[i], c);
    }
}

// ---------------- scan 1: find top bin containing rank k ----------------
__global__ __launch_bounds__(1024) void scan1_kernel(const uint32_t* __restrict__ g,
                                                     uint32_t* __restrict__ ws, uint32_t k) {
    __shared__ uint32_t sums[1024];
    const int t = threadIdx.x;
    const int PER = H1_BINS / 1024;  // 16
    uint32_t local = 0;
    for (int i = 0; i < PER; ++i) local += g[t * PER + i];
    sums[t] = local;
    __syncthreads();
    for (int off = 1; off < 1024; off <<= 1) {
        uint32_t v = (t >= off) ? sums[t - off] : 0u;
        __syncthreads();
        sums[t] += v;
        __syncthreads();
    }
    uint32_t excl = sums[t] - local;
    if (k >= excl && k < excl + local) {
        uint32_t run = excl;
        for (int i = 0; i < PER; ++i) {
            uint32_t c = g[t * PER + i];
            if (k < run + c) { ws[0] = (uint32_t)(t * PER + i); ws[1] = k - run; break; }
            run += c;
        }
    }
}

// ---------------- pass 2: filtered histogram of low 18 key bits ----------------
__global__ __launch_bounds__(1024) void hist2_kernel(const float4* __restrict__ x, long nv,
                                                     const uint32_t* __restrict__ ws,
                                                     uint32_t* __restrict__ g2) {
    const uint32_t b1 = ws[0];
    long stride = (long)gridDim.x * blockDim.x;
    for (long i = (long)blockIdx.x * blockDim.x + threadIdx.x; i < nv; i += stride) {
        float4 v = x[i];
        uint32_t k0 = f2key(v.x); if ((k0 >> H2_BITS) == b1) atomicAdd(&g2[k0 & (H2_BINS - 1u)], 1u);
        uint32_t k1 = f2key(v.y); if ((k1 >> H2_BITS) == b1) atomicAdd(&g2[k1 & (H2_BINS - 1u)], 1u);
        uint32_t k2 = f2key(v.z); if ((k2 >> H2_BITS) == b1) atomicAdd(&g2[k2 & (H2_BINS - 1u)], 1u);
        uint32_t k3 = f2key(v.w); if ((k3 >> H2_BITS) == b1) atomicAdd(&g2[k3 & (H2_BINS - 1u)], 1u);
    }
}

// ---------------- scan 2: exact median value ----------------
__global__ __launch_bounds__(1024) void scan2_kernel(const uint32_t* __restrict__ g,
                                                     uint32_t* __restrict__ ws) {
    __shared__ uint32_t sums[1024];
    const uint32_t b1 = ws[0];
    const uint32_t k  = ws[1];
    const int t = threadIdx.x;
    const int PER = H2_BINS / 1024;  // 256
    uint32_t local = 0;
    for (int i = 0; i < PER; ++i) local += g[t * PER + i];
    sums[t] = local;
    __syncthreads();
    for (int off = 1; off < 1024; off <<= 1) {
        uint32_t v = (t >= off) ? sums[t - off] : 0u;
        __syncthreads();
        sums[t] += v;
        __syncthreads();
    }
    uint32_t excl = sums[t] - local;
    if (k >= excl && k < excl + local) {
        uint32_t run = excl;
        for (int i = 0; i < PER; ++i) {
            uint32_t c = g[t * PER + i];
            if (k < run + c) {
                uint32_t key  = (b1 << H2_BITS) | (uint32_t)(t * PER + i);
                uint32_t bits = (key & 0x80000000u) ? (key ^ 0x80000000u) : ~key;
                ws[2] = bits;  // median as raw float bits
                break;
            }
            run += c;
        }
    }
}

// ---------------- pass 3: threshold + 7x7 maxpool + local-max select ----------------
// Tile: rows grow = row0-3+r (r in [0,38)), cols gcol = col0-4+c (c in [0,136)).
// Column b128 groups are 4-aligned; since 3072 % 4 == 0 every group is fully
// in-image or fully out-of-image, so clamped-address async b128 loads are safe
// (out-of-image cells are overwritten with -inf in the fix-up step).
__global__ __launch_bounds__(256) void nms_kernel(const float* __restrict__ x,
                                                  float* __restrict__ out,
                                                  const uint32_t* __restrict__ ws) {
    __shared__ __align__(16) float tile[LH][LW];   // 38*136*4 = 20672 B (row = 34 x 16B)
    __shared__ __align__(16) float hmax[LH][TW];   // 38*128*4 = 19456 B

    const float med = __uint_as_float(ws[2]);
    const int tid  = threadIdx.x;
    const int col0 = blockIdx.x * TW;
    const int row0 = blockIdx.y * TH;
    const int img  = blockIdx.z;
    const size_t imgOff = (size_t)img * W * W;

    // LDS byte offset of tile base (low 32 bits of flat addr == LDS offset on gfx1250)
    const uint32_t ldsBase = (uint32_t)(uintptr_t)(&tile[0][0]);

    // Async global->LDS b128 copy of the raw tile (clamped addresses). ASYNCcnt-tracked.
    for (int e = tid; e < LH * NG; e += 256) {
        int r = e / NG, g = e - r * NG;
        int gr = row0 + r - HALO;
        int gc = col0 + 4 * g - LHW;                 // 4-aligned group base column
        int grc = min(max(gr, 0), W - 1);
        int gcc = min(max(gc, 0), W - 4);
        uint32_t byteOff = (uint32_t)((imgOff + (size_t)grc * W + (size_t)gcc) * 4u);
        uint32_t lds     = ldsBase + (uint32_t)(r * LW + 4 * g) * 4u;
        asm volatile("global_load_async_to_lds_b128 %0, %1, %2"
                     :: "v"(lds), "v"(byteOff), "s"(x) : "memory");
    }
    asm volatile("s_wait_asynccnt 0x0" ::: "memory");

    // In-LDS fix-up over the SAME groups this thread loaded (own asynccnt already drained):
    // apply median threshold; out-of-image cells -> -inf (SAME pad identity).
    for (int e = tid; e < LH * NG; e += 256) {
        int r = e / NG, g = e - r * NG;
        int gr = row0 + r - HALO;
        int gc = col0 + 4 * g - LHW;
        bool rowValid = (gr >= 0) && (gr < W);
        float4* p4 = (float4*)&tile[r][4 * g];
        float4 v = *p4;
        float vv[4] = { v.x, v.y, v.z, v.w };
#pragma unroll
        for (int j = 0; j < 4; ++j) {
            bool valid = rowValid && (gc + j >= 0) && (gc + j < W);
            float t = (vv[j] >= med) ? vv[j] : 0.0f;
            vv[j] = valid ? t : -INFINITY;
        }
        *p4 = make_float4(vv[0], vv[1], vv[2], vv[3]);
    }
    __syncthreads();

    // Horizontal 7-wide max: output col oc covers tile cols oc+1 .. oc+7
    for (int h = tid; h < LH * TW; h += 256) {
        int r = h >> 7, oc = h & (TW - 1);
        float m = tile[r][oc + 1];
#pragma unroll
        for (int j = 2; j <= 7; ++j) m = fmaxf(m, tile[r][oc + j]);
        hmax[r][oc] = m;
    }
    __syncthreads();

    // Vertical 7-wide max + local-max select, coalesced b32 stores
    for (int o = tid; o < TH * TW; o += 256) {
        int r = o >> 7, oc = o & (TW - 1);
        float p = hmax[r][oc];
#pragma unroll
        for (int j = 1; j < 7; ++j) p = fmaxf(p, hmax[r + j][oc]);
        float t = tile[r + HALO][oc + LHW];
        out[imgOff + (size_t)(row0 + r) * W + (size_t)(col0 + oc)] = (t == p) ? p : 0.0f;
    }
}

extern "C" void kernel_launch(void* const* d_in, const int* in_sizes, int n_in,
                              void* d_out, int out_size, void* d_ws, size_t ws_size,
                              hipStream_t stream) {
    const float*  x   = (const float*)d_in[0];
    const float4* xv  = (const float4*)d_in[0];
    float*        out = (float*)d_out;
    uint32_t*     ws  = (uint32_t*)d_ws;

    const long n  = (long)in_sizes[0];          // 8*3072*3072
    const long nv = n / 4;
    const uint32_t k = (uint32_t)((n - 1) / 2); // lower-median rank

    // Histogram scratch lives in d_out (fully overwritten by nms_kernel at the end).
    uint32_t* hist1 = (uint32_t*)d_out;
    uint32_t* hist2 = hist1 + H1_BINS;
    const int nzero = (int)(H1_BINS + H2_BINS);

    zero_kernel<<<dim3((nzero + 255) / 256), dim3(256), 0, stream>>>(hist1, ws, nzero);
    hist1_kernel<<<dim3(1024), dim3(1024), 0, stream>>>(xv, nv, hist1);
    scan1_kernel<<<dim3(1), dim3(1024), 0, stream>>>(hist1, ws, k);
    hist2_kernel<<<dim3(1024), dim3(1024), 0, stream>>>(xv, nv, ws, hist2);
    scan2_kernel<<<dim3(1), dim3(1024), 0, stream>>>(hist2, ws);

    dim3 grid(W / TW, W / TH, 8);  // 24 x 96 x 8
    nms_kernel<<<grid, dim3(256), 0, stream>>>(x, out, ws);
}